// NNUE_53352083751150
// MI455X (gfx1250) — compile-verified
//
#include <hip/hip_runtime.h>

typedef __attribute__((ext_vector_type(2))) float v2f;
typedef __attribute__((ext_vector_type(8))) float v8f;

#define NUM_FEATURES 40960
#define BATCH        8192
#define ROWS         16            // batch rows per workgroup
#define KC           256           // K chunk staged in LDS per buffer
#define PITCH        260           // LDS row pitch in floats (mod 64 = 4 -> conflict-free, 16B aligned)
#define WAVES        8
#define KW           (KC / WAVES)  // 32 K columns per wave per chunk
#define NCHUNK       (NUM_FEATURES / KC)   // 160

// ---- CDNA5 async global->LDS copy (bypasses VGPRs, tracked on ASYNCcnt) ----
__device__ __forceinline__ void async_b128(unsigned lds_byte_off, const float* gaddr) {
    asm volatile("global_load_async_to_lds_b128 %0, %1, off"
                 :: "v"(lds_byte_off), "v"(gaddr) : "memory");
}
__device__ __forceinline__ void wait_async_le9() {
#if __has_builtin(__builtin_amdgcn_s_wait_asynccnt)
    __builtin_amdgcn_s_wait_asynccnt(9);
#else
    asm volatile("s_wait_asynccnt 0x9" ::: "memory");
#endif
}
__device__ __forceinline__ void wait_async_0() {
#if __has_builtin(__builtin_amdgcn_s_wait_asynccnt)
    __builtin_amdgcn_s_wait_asynccnt(0);
#else
    asm volatile("s_wait_asynccnt 0x0" ::: "memory");
#endif
}

__launch_bounds__(256)
__global__ void nnue_ft_kernel(const float* __restrict__ wfts,
                               const float* __restrict__ bfts,
                               const float* __restrict__ stm,
                               const float* __restrict__ ft_w,
                               const float* __restrict__ ft_b,
                               const float* __restrict__ l1_w,
                               const float* __restrict__ l1_b,
                               const float* __restrict__ l2_w,
                               const float* __restrict__ l2_b,
                               const float* __restrict__ l3_w,
                               const float* __restrict__ l3_b,
                               float* __restrict__ out)
{
    __shared__ float lds_w[2][ROWS * PITCH];   // wfts tiles, double-buffered
    __shared__ float lds_b[2][ROWS * PITCH];   // bfts tiles
    __shared__ float lds_f[2][5 * PITCH];      // ft_w rows 0-3 + zero row 4
    __shared__ float red_w[WAVES][ROWS][4];
    __shared__ float red_b[WAVES][ROWS][4];

    const int tid  = threadIdx.x;
    const int wave = tid >> 5;
    const int lane = tid & 31;
    const int half = lane >> 4;        // 0: lanes 0-15, 1: lanes 16-31
    const int l16  = lane & 15;
    const int brow = (l16 < 4) ? l16 : 4;   // ft_w row (row 4 = zeros for unused N)
    const int row0 = blockIdx.x * ROWS;

    // zero row 4 of both ft_w buffers (lanes n>=4 broadcast-read it)
    lds_f[0][4 * PITCH + tid] = 0.f;
    lds_f[1][4 * PITCH + tid] = 0.f;

    v8f cw = {0.f, 0.f, 0.f, 0.f, 0.f, 0.f, 0.f, 0.f};
    v8f cb = {0.f, 0.f, 0.f, 0.f, 0.f, 0.f, 0.f, 0.f};

    // ---- async-stage one chunk (9 x b128 per thread: 4 wfts + 4 bfts + 1 ft_w) ----
    auto stage = [&](int kbase, int buf) {
        const unsigned lw = (unsigned)(uintptr_t)&lds_w[buf][0];
        const unsigned lb = (unsigned)(uintptr_t)&lds_b[buf][0];
        const unsigned lf = (unsigned)(uintptr_t)&lds_f[buf][0];
        #pragma unroll
        for (int i = 0; i < 4; ++i) {
            int q  = i * 256 + tid;        // quad index; 64 quads per row
            int r  = q >> 6;
            int c4 = q & 63;
            unsigned lofs = (unsigned)((r * PITCH + c4 * 4) * 4);
            long gidx = (long)(row0 + r) * NUM_FEATURES + kbase + c4 * 4;
            async_b128(lw + lofs, wfts + gidx);
            async_b128(lb + lofs, bfts + gidx);
        }
        {
            int r  = tid >> 6;             // ft_w rows 0-3
            int c4 = tid & 63;
            unsigned lofs = (unsigned)((r * PITCH + c4 * 4) * 4);
            async_b128(lf + lofs, ft_w + (long)r * NUM_FEATURES + kbase + c4 * 4);
        }
    };

    stage(0, 0);   // prologue: chunk 0 -> buffer 0

    for (int c = 0; c < NCHUNK; ++c) {
        const int buf = c & 1;
        if (c + 1 < NCHUNK) {
            stage((c + 1) * KC, buf ^ 1);  // prefetch next chunk into other buffer
            wait_async_le9();              // <=9 outstanding => chunk c fully in LDS
        } else {
            wait_async_0();
        }
        __syncthreads();

        // ---- branch-free inner loop: 3x ds_load_b64 + 2x v_wmma per k-step ----
        const float* BW = lds_w[buf];
        const float* BB = lds_b[buf];
        const float* BF = lds_f[buf];
        const int kcol0 = wave * KW;
        #pragma unroll
        for (int kk = 0; kk < KW; kk += 4) {
            const int kc = kcol0 + kk;
            v2f aw = *(const v2f*)(&BW[l16  * PITCH + kc + 2 * half]);
            v2f ab = *(const v2f*)(&BB[l16  * PITCH + kc + 2 * half]);
            v2f bt = *(const v2f*)(&BF[brow * PITCH + kc + 2 * half]);
            cw = __builtin_amdgcn_wmma_f32_16x16x4_f32(false, aw, false, bt,
                                                       (short)0, cw, false, false);
            cb = __builtin_amdgcn_wmma_f32_16x16x4_f32(false, ab, false, bt,
                                                       (short)0, cb, false, false);
        }
        __syncthreads();   // all reads of buf done before it is overwritten
    }

    // ---- cross-wave reduction of the 4 useful output columns ----
    // D layout: lane l16 = column n, VGPR v -> row m = v + 8*half
    if (l16 < 4) {
        #pragma unroll
        for (int v = 0; v < 8; ++v) {
            red_w[wave][v + 8 * half][l16] = cw[v];
            red_b[wave][v + 8 * half][l16] = cb[v];
        }
    }
    __syncthreads();

    // ---- tail: stm mux + crelu + 8x8 -> 8x8 -> 8x1 MLP, one thread per row ----
    if (tid < ROWS) {
        const int m = tid;
        float w[4], b[4];
        #pragma unroll
        for (int n = 0; n < 4; ++n) {
            float sw = ft_b[n], sb = ft_b[n];
            #pragma unroll
            for (int wv = 0; wv < WAVES; ++wv) {
                sw += red_w[wv][m][n];
                sb += red_b[wv][m][n];
            }
            w[n] = sw; b[n] = sb;
        }
        const float s = stm[row0 + m];
        float acc[8], x1[8], x2[8];
        #pragma unroll
        for (int j = 0; j < 4; ++j) {
            acc[j]     = s * w[j] + (1.f - s) * b[j];
            acc[j + 4] = s * b[j] + (1.f - s) * w[j];
        }
        #pragma unroll
        for (int j = 0; j < 8; ++j) acc[j] = fminf(fmaxf(acc[j], 0.f), 1.f);
        #pragma unroll
        for (int i = 0; i < 8; ++i) {
            float t = l1_b[i];
            #pragma unroll
            for (int j = 0; j < 8; ++j) t += acc[j] * l1_w[i * 8 + j];
            x1[i] = fminf(fmaxf(t, 0.f), 1.f);
        }
        #pragma unroll
        for (int i = 0; i < 8; ++i) {
            float t = l2_b[i];
            #pragma unroll
            for (int j = 0; j < 8; ++j) t += x1[j] * l2_w[i * 8 + j];
            x2[i] = fminf(fmaxf(t, 0.f), 1.f);
        }
        float o = l3_b[0];
        #pragma unroll
        for (int j = 0; j < 8; ++j) o += x2[j] * l3_w[j];
        out[row0 + m] = o;
    }
}

extern "C" void kernel_launch(void* const* d_in, const int* in_sizes, int n_in,
                              void* d_out, int out_size, void* d_ws, size_t ws_size,
                              hipStream_t stream) {
    const float* wfts = (const float*)d_in[0];
    const float* bfts = (const float*)d_in[1];
    const float* stm  = (const float*)d_in[2];
    const float* ft_w = (const float*)d_in[3];
    const float* ft_b = (const float*)d_in[4];
    const float* l1_w = (const float*)d_in[5];
    const float* l1_b = (const float*)d_in[6];
    const float* l2_w = (const float*)d_in[7];
    const float* l2_b = (const float*)d_in[8];
    const float* l3_w = (const float*)d_in[9];
    const float* l3_b = (const float*)d_in[10];
    float* out = (float*)d_out;

    dim3 grid(BATCH / ROWS);   // 512 workgroups, 16 batch rows each
    dim3 block(256);           // 8 wave32s
    nnue_ft_kernel<<<grid, block, 0, stream>>>(wfts, bfts, stm, ft_w, ft_b,
                                               l1_w, l1_b, l2_w, l2_b,
                                               l3_w, l3_b, out);
}